// AttentionInformer_72215580115005
// MI455X (gfx1250) — compile-verified
//
#include <hip/hip_runtime.h>
#include <float.h>

#define EMB   1024
#define NROW  65536
#define PICK  12
#define LDSR  1028   // padded LDS row stride (floats): stride%64==4 -> conflict-free b64 reads

typedef float v2f __attribute__((ext_vector_type(2)));
typedef float v8f __attribute__((ext_vector_type(8)));

__global__ __launch_bounds__(256) void zero_kernel(float* __restrict__ p, int n) {
  int i = blockIdx.x * blockDim.x + threadIdx.x;
  if (i < n) p[i] = 0.0f;
}

// rows_out[p,k] = dot(embed[idx[p],:], W[k,:]) + b[k]   for p<12, k<1024
__global__ __launch_bounds__(128) void rows_linear_kernel(
    const float* __restrict__ embed, const float* __restrict__ W,
    const float* __restrict__ b, const int* __restrict__ idx,
    float* __restrict__ rows_out) {
  __shared__ float er[PICK][EMB];   // 48 KB
  const int tid = threadIdx.x;
  for (int p = 0; p < PICK; ++p) {
    const size_t r = (size_t)idx[p];
    for (int j = tid; j < EMB; j += 128) er[p][j] = embed[r * EMB + j];
  }
  __syncthreads();
  const int k = blockIdx.x * 128 + tid;
  const float* wr = W + (size_t)k * EMB;
  float acc[PICK];
  #pragma unroll
  for (int p = 0; p < PICK; ++p) acc[p] = 0.0f;
  for (int j = 0; j < EMB; j += 4) {
    float4 w4 = *(const float4*)(wr + j);
    #pragma unroll
    for (int p = 0; p < PICK; ++p) {
      acc[p] = fmaf(er[p][j + 0], w4.x, acc[p]);
      acc[p] = fmaf(er[p][j + 1], w4.y, acc[p]);
      acc[p] = fmaf(er[p][j + 2], w4.z, acc[p]);
      acc[p] = fmaf(er[p][j + 3], w4.w, acc[p]);
    }
  }
  const float bk = b[k];
  for (int p = 0; p < PICK; ++p) rows_out[p * EMB + k] = acc[p] + bk;
}

// Aout[p,j] = sum_k rows[p,k] * W2[k,j];  cvec[p] = rows[p,:]·b2 (p<12), cvec[12..15] = -3e38
__global__ __launch_bounds__(256) void proj_kernel(
    const float* __restrict__ rows, const float* __restrict__ W2,
    const float* __restrict__ b2, float* __restrict__ Aout,
    float* __restrict__ cvec) {
  __shared__ float rl[PICK][EMB];   // 48 KB
  const int tid = threadIdx.x;
  for (int p = 0; p < PICK; ++p)
    for (int j = tid; j < EMB; j += 256) rl[p][j] = rows[p * EMB + j];
  __syncthreads();
  const int j = blockIdx.x * 256 + tid;
  float acc[PICK];
  #pragma unroll
  for (int p = 0; p < PICK; ++p) acc[p] = 0.0f;
  for (int k = 0; k < EMB; ++k) {
    const float w = W2[(size_t)k * EMB + j];   // coalesced over j
    #pragma unroll
    for (int p = 0; p < PICK; ++p) acc[p] = fmaf(rl[p][k], w, acc[p]);
  }
  for (int p = 0; p < PICK; ++p) Aout[p * EMB + j] = acc[p];
  if (blockIdx.x == 0 && tid < 16) {
    if (tid < PICK) {
      float s = 0.0f;
      for (int k = 0; k < EMB; ++k) s = fmaf(rl[tid][k], b2[k], s);
      cvec[tid] = s;
    } else {
      cvec[tid] = -3.0e38f;   // padded columns can never win the max
    }
  }
}

// Per 16-row tile of embed: D[16n x 16p] = embed_tile @ Amat.T + c  (WMMA f32 16x16x4),
// then pooled[n] = max_p D[n,p]. Pass1: store pooled. Pass2: out += pooled[n]*embed[n,:].
__global__ __launch_bounds__(256) void pass_kernel(
    const float* __restrict__ embed, const float* __restrict__ Amat,
    const float* __restrict__ cvec, float* __restrict__ max_out,
    float* __restrict__ accum_out, const int do_accum) {
  __shared__ float Bl[13 * LDSR];   // 12 data rows + 1 zero row (for pad lanes 12..15)
  __shared__ float accl[EMB];
  const int tid = threadIdx.x;
  for (int p = 0; p < PICK; ++p)
    for (int j = tid; j < EMB; j += 256) Bl[p * LDSR + j] = Amat[p * EMB + j];
  for (int j = tid; j < EMB; j += 256) Bl[PICK * LDSR + j] = 0.0f;
  if (do_accum)
    for (int j = tid; j < EMB; j += 256) accl[j] = 0.0f;
  __syncthreads();

  const int lane = tid & 31, wave = tid >> 5;
  const int l16 = lane & 15, half = lane >> 4;
  const int prow = (l16 < PICK) ? l16 : PICK;   // pad lanes read the zero row (broadcast)
  const size_t rowbase = ((size_t)blockIdx.x * 8 + wave) * 16;

  // C init = c[p] (bias fold); padded p start at -3e38 and accumulate zeros.
  const float cv = cvec[l16];
  v8f acc;
  #pragma unroll
  for (int i = 0; i < 8; ++i) acc[i] = cv;

  // A frag: lane supplies embed[row=rowbase+l16][k0 + 2*half + {0,1}]
  // B frag: lane supplies Amat[p=l16][k0 + 2*half + {0,1}]  (column-major B per ISA layout)
  const float* aptr = embed + (rowbase + (size_t)l16) * EMB + 2 * half;
  const float* bptr = &Bl[prow * LDSR + 2 * half];

#if __has_builtin(__builtin_amdgcn_wmma_f32_16x16x4_f32)
  #pragma unroll 4
  for (int k0 = 0; k0 < EMB; k0 += 4) {
    v2f a  = *(const v2f*)(aptr + k0);
    v2f bb = *(const v2f*)(bptr + k0);
    acc = __builtin_amdgcn_wmma_f32_16x16x4_f32(false, a, false, bb,
                                                (short)0, acc, false, false);
  }
#else
  for (int k0 = 0; k0 < EMB; ++k0) {   // VALU fallback, same semantics
    const float bv = Bl[prow * LDSR + k0];
    #pragma unroll
    for (int r = 0; r < 8; ++r)
      acc[r] = fmaf(embed[(rowbase + (size_t)(r + 8 * half)) * EMB + k0], bv, acc[r]);
  }
#endif

  // D layout: vgpr r, lane L -> (m = r + 8*(L/16), p = L%16). Row-max over p via
  // 16-lane butterfly (xor 1,2,4,8 stays within each half).
  float pm[8];
  #pragma unroll
  for (int r = 0; r < 8; ++r) {
    float v = acc[r];
    v = fmaxf(v, __shfl_xor(v, 1, 32));
    v = fmaxf(v, __shfl_xor(v, 2, 32));
    v = fmaxf(v, __shfl_xor(v, 4, 32));
    v = fmaxf(v, __shfl_xor(v, 8, 32));
    pm[r] = v;   // pooled[rowbase + r + 8*half], same in all lanes of the half
  }

  if (!do_accum) {
    if (l16 == 0) {
      #pragma unroll
      for (int r = 0; r < 8; ++r) max_out[rowbase + r + 8 * half] = pm[r];
    }
  } else {
    // out[j] += sum_m pooled[m] * embed[m][j]; halves cover rows 0-7 / 8-15.
    for (int jb = 0; jb < EMB; jb += 16) {
      const int j = jb + l16;
      float s = 0.0f;
      #pragma unroll
      for (int r = 0; r < 8; ++r)
        s = fmaf(pm[r], embed[(rowbase + (size_t)(r + 8 * half)) * EMB + j], s);
      atomicAdd(&accl[j], s);            // ds_add_f32
    }
    __syncthreads();
    for (int j = tid; j < EMB; j += 256) atomicAdd(&accum_out[j], accl[j]);
  }
}

__global__ __launch_bounds__(256) void topk_kernel(float* __restrict__ vals,
                                                   int* __restrict__ top_idx) {
  __shared__ float sv[256];
  __shared__ int si[256];
  const int tid = threadIdx.x;
  for (int round = 0; round < PICK; ++round) {
    float best = -FLT_MAX;
    int bi = NROW;
    for (int i = tid; i < NROW; i += 256) {
      const float v = vals[i];
      if (v > best) { best = v; bi = i; }   // keeps lowest index on ties within slice
    }
    sv[tid] = best; si[tid] = bi;
    __syncthreads();
    for (int s = 128; s > 0; s >>= 1) {
      if (tid < s) {
        const float vo = sv[tid + s]; const int io = si[tid + s];
        if (vo > sv[tid] || (vo == sv[tid] && io < si[tid])) { sv[tid] = vo; si[tid] = io; }
      }
      __syncthreads();
    }
    if (tid == 0) { top_idx[round] = si[0]; vals[si[0]] = -FLT_MAX; }
    __syncthreads();
  }
}

extern "C" void kernel_launch(void* const* d_in, const int* in_sizes, int n_in,
                              void* d_out, int out_size, void* d_ws, size_t ws_size,
                              hipStream_t stream) {
  (void)in_sizes; (void)n_in; (void)out_size; (void)ws_size;
  const float* embed   = (const float*)d_in[0];
  const float* Wq      = (const float*)d_in[1];
  const float* bq      = (const float*)d_in[2];
  const float* Wk      = (const float*)d_in[3];
  const float* bk      = (const float*)d_in[4];
  const int*   indices = (const int*)d_in[5];
  float* out = (float*)d_out;
  float* ws  = (float*)d_ws;

  float* new_key = ws;                   // 12*1024
  float* Amat    = ws + 12288;           // 12*1024
  float* cvec    = ws + 24576;           // 16
  float* Qred    = ws + 24592;           // 12*1024
  float* Rmat    = ws + 36880;           // 12*1024
  float* dvec    = ws + 49168;           // 16
  float* maxv    = ws + 49184;           // 65536
  int*   tidx    = (int*)(ws + 114720);  // 12 ints (~459 KB total workspace)

  zero_kernel<<<4, 256, 0, stream>>>(out, EMB);
  // new_key = embed[indices] @ Wk.T + bk
  rows_linear_kernel<<<8, 128, 0, stream>>>(embed, Wk, bk, indices, new_key);
  // A = new_key @ Wq ; c = new_key @ bq
  proj_kernel<<<4, 256, 0, stream>>>(new_key, Wq, bq, Amat, cvec);
  // max_values[n] = max_p (embed[n]·A[p] + c[p])   (streaming pass 1, WMMA)
  pass_kernel<<<512, 256, 0, stream>>>(embed, Amat, cvec, maxv, nullptr, 0);
  topk_kernel<<<1, 256, 0, stream>>>(maxv, tidx);
  // Q_reduced = embed[top] @ Wq.T + bq
  rows_linear_kernel<<<8, 128, 0, stream>>>(embed, Wq, bq, tidx, Qred);
  // R = Q_reduced @ Wk ; d = Q_reduced @ bk
  proj_kernel<<<4, 256, 0, stream>>>(Qred, Wk, bk, Rmat, dvec);
  // pooled[n] = max_p (embed[n]·R[p] + d[p]); out += pooled[n]*embed[n]  (pass 2, fused)
  pass_kernel<<<512, 256, 0, stream>>>(embed, Rmat, dvec, nullptr, out, 1);
}